// JacobiMachine_51135880626180
// MI455X (gfx1250) — compile-verified
//
#include <hip/hip_runtime.h>
#include <hip/hip_bf16.h>
#include <cstdint>

// Reference reduces (after DCE) to: out = jacobi(x0, Mask1, 100) with
//   x0 = exp(-50((X-.5)^2+(Y-.5)^2)),  step: x = 0.25*(4-neighbor sum, zero pad) * Mask1
#define HGRID 1024
#define BI    64          // interior block per workgroup
#define HALO  8           // steps per pass
#define TD    80          // BI + 2*HALO, tile dim (5 x 16 -> WMMA friendly)
#define TDS   81          // padded LDS row stride (avoid 64-bank conflicts)
#define NBLK  5           // TD / 16
#define NTH   160         // 5 wave32s: one wave per round of 5 WMMA blocks

typedef float v2f __attribute__((ext_vector_type(2)));
typedef float v8f __attribute__((ext_vector_type(8)));

// CDNA5 async copy: global -> LDS, tracked by ASYNCcnt (CDNA5 ISA)
__device__ __forceinline__ void async_load_f32(const float* g, const float* lds_ptr) {
  unsigned loff = (unsigned)(uintptr_t)lds_ptr;  // low 32b of flat shared addr == LDS offset
  asm volatile("global_load_async_to_lds_b32 %0, %1, off"
               :: "v"(loff), "v"(g) : "memory");
}
__device__ __forceinline__ void wait_async_all() {
  asm volatile("s_wait_asynccnt 0" ::: "memory");
}

__global__ __launch_bounds__(256) void init_x0(const float* __restrict__ X,
                                               const float* __restrict__ Y,
                                               float* __restrict__ out) {
  int i = blockIdx.x * blockDim.x + threadIdx.x;
  float dx = X[i] - 0.5f;
  float dy = Y[i] - 0.5f;
  out[i] = expf(-50.f * (dx * dx + dy * dy));
}

// One temporal-blocked pass: load (BI+2*HALO)^2 tile into LDS via async copies,
// run `steps` (<= HALO) Jacobi iterations in LDS using WMMA (stencil-as-matmul
// with tridiagonal shift matrix T: D = T*X + X*T + halo edge fixups), write the
// BI x BI interior back to global.
__global__ __launch_bounds__(NTH) void jacobi_pass(const float* __restrict__ src,
                                                   const float* __restrict__ mask,
                                                   float* __restrict__ dst,
                                                   int steps) {
  __shared__ float tA[TD * TDS];
  __shared__ float tB[TD * TDS];
  __shared__ float tM[TD * TDS];

  const int tid = threadIdx.x;
  const int bx = blockIdx.x, by = blockIdx.y;
  const int gr0 = by * BI - HALO;
  const int gc0 = bx * BI - HALO;

  // zero-fill (out-of-domain cells stay 0; mask 0 keeps them 0 every step)
  for (int i = tid; i < TD * TDS; i += NTH) { tA[i] = 0.f; tM[i] = 0.f; }
  __syncthreads();

  // async stage x-tile and mask-tile into LDS (in-domain lanes only)
  for (int i = tid; i < TD * TD; i += NTH) {
    int r = i / TD, c = i - r * TD;
    int gr = gr0 + r, gc = gc0 + c;
    if (gr >= 0 && gr < HGRID && gc >= 0 && gc < HGRID) {
      const float* gp = src  + gr * HGRID + gc;
      const float* mp = mask + gr * HGRID + gc;
      async_load_f32(gp, &tA[r * TDS + c]);
      async_load_f32(mp, &tM[r * TDS + c]);
    }
  }
  wait_async_all();
  __syncthreads();

  float* cur = tA;
  float* nxt = tB;
  const int wave = tid >> 5;
  const int lane = tid & 31;
  const int n    = lane & 15;   // column (B/C/D layout) and row m (A layout)
  const int half = lane >> 4;   // lane group 0/1

  for (int it = 0; it < steps; ++it) {
    for (int blk = wave; blk < NBLK * NBLK; blk += NTH / 32) {
      const int bi = blk / NBLK, bj = blk - bi * NBLK;
      const int r0 = bi * 16, c0 = bj * 16;
      v8f d = {0.f, 0.f, 0.f, 0.f, 0.f, 0.f, 0.f, 0.f};

      // D = T*X + X*T   (T tridiagonal 0/1; K=16 via 4 chained 16x16x4 WMMAs each)
#pragma unroll
      for (int kb = 0; kb < 16; kb += 4) {
        const int ks = kb + 2 * half;
        v2f tS;  // per-lane slice of T: A-layout == B-layout (T symmetric)
        tS.x = (n == ks - 1 || n == ks + 1) ? 1.f : 0.f;  // T[n][ks]
        tS.y = (n == ks     || n == ks + 2) ? 1.f : 0.f;  // T[n][ks+1]
        v2f bX;  // B-operand: rows ks,ks+1 of X
        bX.x = cur[(r0 + ks)     * TDS + c0 + n];
        bX.y = cur[(r0 + ks + 1) * TDS + c0 + n];
        d = __builtin_amdgcn_wmma_f32_16x16x4_f32(false, tS, false, bX,
                                                  (short)0, d, false, false);
        v2f aX;  // A-operand: cols ks,ks+1 of X (row = n)
        aX.x = cur[(r0 + n) * TDS + c0 + ks];
        aX.y = cur[(r0 + n) * TDS + c0 + ks + 1];
        d = __builtin_amdgcn_wmma_f32_16x16x4_f32(false, aX, false, tS,
                                                  (short)0, d, false, false);
      }

      // cross-block halo fixups (rows M==0 / M==15 of the block)
      {
        int ru = (bi > 0) ? (r0 - 1) : 0;
        float up = cur[ru * TDS + c0 + n];
        d[0] += (half == 0 && bi > 0) ? up : 0.f;          // element M==0
        int rd = (bi < NBLK - 1) ? (r0 + 16) : 0;
        float dn = cur[rd * TDS + c0 + n];
        d[7] += (half == 1 && bi < NBLK - 1) ? dn : 0.f;   // element M==15
      }
      // cross-block halo fixups (cols N==0 / N==15)
      if (n == 0 && bj > 0) {
#pragma unroll
        for (int v = 0; v < 8; ++v) {
          int r = r0 + v + 8 * half;
          d[v] += cur[r * TDS + c0 - 1];
        }
      }
      if (n == 15 && bj < NBLK - 1) {
#pragma unroll
        for (int v = 0; v < 8; ++v) {
          int r = r0 + v + 8 * half;
          d[v] += cur[r * TDS + c0 + 16];
        }
      }

      // new = ring ? old : 0.25*D*mask  (tile ring kept stale; interior >= HALO away)
#pragma unroll
      for (int v = 0; v < 8; ++v) {
        int r = r0 + v + 8 * half;
        int c = c0 + n;
        float oldv = cur[r * TDS + c];
        float mv   = tM[r * TDS + c];
        bool ring = (r == 0) | (r == TD - 1) | (c == 0) | (c == TD - 1);
        nxt[r * TDS + c] = ring ? oldv : (0.25f * d[v] * mv);
      }
    }
    __syncthreads();
    float* t = cur; cur = nxt; nxt = t;
  }

  // write back interior BI x BI
  for (int i = tid; i < BI * BI; i += NTH) {
    int r = i >> 6, c = i & (BI - 1);
    dst[(by * BI + r) * HGRID + (bx * BI + c)] = cur[(r + HALO) * TDS + (c + HALO)];
  }
}

extern "C" void kernel_launch(void* const* d_in, const int* in_sizes, int n_in,
                              void* d_out, int out_size, void* d_ws, size_t ws_size,
                              hipStream_t stream) {
  const float* X  = (const float*)d_in[0];
  const float* Y  = (const float*)d_in[1];
  const float* M1 = (const float*)d_in[2];   // Mask1: 1x1x1024x1024
  float* out = (float*)d_out;
  float* wsA = (float*)d_ws;                 // 4 MB scratch; ping-pong with d_out

  (void)in_sizes; (void)n_in; (void)out_size; (void)ws_size;

  init_x0<<<(HGRID * HGRID) / 256, 256, 0, stream>>>(X, Y, wsA);

  dim3 grid(HGRID / BI, HGRID / BI);
  int done = 0, pass = 0;
  while (done < 100) {
    int steps = (100 - done) < HALO ? (100 - done) : HALO;  // 12x8 + 4
    const float* s = (pass & 1) ? (const float*)out : (const float*)wsA;
    float* dptr    = (pass & 1) ? wsA : out;                // last pass (12, even) -> out
    jacobi_pass<<<grid, NTH, 0, stream>>>(s, M1, dptr, steps);
    done += steps;
    ++pass;
  }
}